// Memory_4698694221967
// MI455X (gfx1250) — compile-verified
//
#include <hip/hip_runtime.h>

typedef float v2f __attribute__((ext_vector_type(2)));
typedef float v8f __attribute__((ext_vector_type(8)));

#define BB   128
#define LL   1024
#define DK   128
#define DV   256
#define NN   50
#define HH   256
#define ROWS (BB * LL)
#define KCAT (DV + DK)   // 384

// Fast activations: CDNA5 has V_TANH_F32 (transcendental VALU).
__device__ __forceinline__ float fast_tanh(float x) {
#if __has_builtin(__builtin_amdgcn_tanhf)
  return __builtin_amdgcn_tanhf(x);
#else
  return tanhf(x);
#endif
}

// ---------------------------------------------------------------------------
// Kernel 1: gates. e = sigmoid(I @ We^T + be), a = tanh(I @ Wa^T + ba)
// One wave per 32x64 output tile (2 A-frags x 4 B-frags = 8 accumulators):
// steady state = 6 loads : 8 WMMAs. FP32 WMMA 16x16x4, K = 256.
// Activation is uniform: act(v) = s*tanh(s*v) + o, (s,o)=(1,0) tanh,
// (0.5,0.5) sigmoid -> exactly one v_tanh_f32 per element, no dual paths.
// ---------------------------------------------------------------------------
__global__ __launch_bounds__(256) void gates_kernel(
    const float* __restrict__ X,                         // (ROWS, DV)
    const float* __restrict__ We, const float* __restrict__ be,
    const float* __restrict__ Wa, const float* __restrict__ ba,
    float* __restrict__ e_out, float* __restrict__ a_out)
{
  const int wave = blockIdx.x * 8 + (threadIdx.x >> 5);
  const int lane = threadIdx.x & 31;
  const int which = wave & 1;            // 0 -> erase/sigmoid, 1 -> add/tanh
  const int ct4   = (wave >> 1) & 3;     // 4 strips of 64 cols
  const int rt    = wave >> 3;           // 4096 row tiles of 32

  const float* W    = which ? Wa : We;
  const float* bias = which ? ba : be;
  float*       out  = which ? a_out : e_out;
  const float  s_act = which ? 1.0f : 0.5f;
  const float  o_act = which ? 0.0f : 0.5f;

  const int r0 = rt * 32, c0 = ct4 * 64;
  const int mn = lane & 15;              // row of A frag / col of B frag
  const int kh = ((lane >> 4) & 1) * 2;  // K sub-offset per lane half

  const float* arow = X + (size_t)(r0 + mn) * DV + kh;   // rows r0..r0+15
  const float* brow = W + (size_t)(c0 + mn) * DV + kh;   // B[k][n] = W[n][k]
  __builtin_prefetch(arow, 0, 3);
  __builtin_prefetch(brow, 0, 3);

  v8f acc[2][4] = {};
#pragma unroll 2
  for (int k = 0; k < DV; k += 4) {
    v2f a0 = *(const v2f*)(arow + k);                    // rows r0+0..15
    v2f a1 = *(const v2f*)(arow + 16 * DV + k);          // rows r0+16..31
    v2f b0 = *(const v2f*)(brow + 0 * 16 * DV + k);
    v2f b1 = *(const v2f*)(brow + 1 * 16 * DV + k);
    v2f b2 = *(const v2f*)(brow + 2 * 16 * DV + k);
    v2f b3 = *(const v2f*)(brow + 3 * 16 * DV + k);
    acc[0][0] = __builtin_amdgcn_wmma_f32_16x16x4_f32(false, a0, false, b0, (short)0, acc[0][0], false, false);
    acc[0][1] = __builtin_amdgcn_wmma_f32_16x16x4_f32(false, a0, false, b1, (short)0, acc[0][1], false, false);
    acc[0][2] = __builtin_amdgcn_wmma_f32_16x16x4_f32(false, a0, false, b2, (short)0, acc[0][2], false, false);
    acc[0][3] = __builtin_amdgcn_wmma_f32_16x16x4_f32(false, a0, false, b3, (short)0, acc[0][3], false, false);
    acc[1][0] = __builtin_amdgcn_wmma_f32_16x16x4_f32(false, a1, false, b0, (short)0, acc[1][0], false, false);
    acc[1][1] = __builtin_amdgcn_wmma_f32_16x16x4_f32(false, a1, false, b1, (short)0, acc[1][1], false, false);
    acc[1][2] = __builtin_amdgcn_wmma_f32_16x16x4_f32(false, a1, false, b2, (short)0, acc[1][2], false, false);
    acc[1][3] = __builtin_amdgcn_wmma_f32_16x16x4_f32(false, a1, false, b3, (short)0, acc[1][3], false, false);
  }

#pragma unroll
  for (int rr = 0; rr < 2; ++rr) {
    const int rbase = r0 + rr * 16 + ((lane >> 4) & 1) * 8;
#pragma unroll
    for (int cc = 0; cc < 4; ++cc) {
      const int col = c0 + cc * 16 + mn;
      const float bvv = bias[col];
#pragma unroll
      for (int i = 0; i < 8; ++i) {
        float v = acc[rr][cc][i] + bvv;
        v = fmaf(s_act, fast_tanh(s_act * v), o_act);
        out[(size_t)(rbase + i) * DV + col] = v;
      }
    }
  }
}

// ---------------------------------------------------------------------------
// Kernel 2: score = softmax(problems @ w_key^T) over N=50.
// One thread per (b,l) row; w_key staged in LDS; 50 accumulators in VGPRs.
// ---------------------------------------------------------------------------
__global__ __launch_bounds__(256) void score_kernel(
    const float* __restrict__ P,    // (ROWS, DK)
    const float* __restrict__ WK,   // (NN, DK)
    float* __restrict__ S)          // (ROWS, NN)
{
  __shared__ float wk[NN * DK];     // 25.6 KB
  for (int i = threadIdx.x; i < NN * DK; i += 256) wk[i] = WK[i];
  __syncthreads();

  const size_t row = (size_t)blockIdx.x * 256 + threadIdx.x;
  const float* pr = P + row * DK;

  float sc[NN];
#pragma unroll
  for (int n = 0; n < NN; ++n) sc[n] = 0.0f;

  for (int kc = 0; kc < DK; kc += 8) {
    float pk[8];
#pragma unroll
    for (int j = 0; j < 8; ++j) pk[j] = pr[kc + j];
#pragma unroll
    for (int n = 0; n < NN; ++n) {
#pragma unroll
      for (int j = 0; j < 8; ++j)
        sc[n] = fmaf(pk[j], wk[n * DK + kc + j], sc[n]);
    }
  }

  float mx = sc[0];
#pragma unroll
  for (int n = 1; n < NN; ++n) mx = fmaxf(mx, sc[n]);
  float sum = 0.0f;
#pragma unroll
  for (int n = 0; n < NN; ++n) { sc[n] = __expf(sc[n] - mx); sum += sc[n]; }
  const float inv = 1.0f / sum;
  float* so = S + row * NN;
#pragma unroll
  for (int n = 0; n < NN; ++n) so[n] = sc[n] * inv;
}

// ---------------------------------------------------------------------------
// Kernel 3: sequential scan. Block = (batch, 64-wide d-chunk), 128 threads.
// Thread = (d = tid>>1, n-half = tid&1): halves sit on ADJACENT lanes so the
// read reduction is one __shfl_xor (no barrier). Memory state (25 elems per
// thread) lives in registers. Scores double-buffered: 1 barrier per step.
// ---------------------------------------------------------------------------
__global__ __launch_bounds__(128) void scan_kernel(
    const float* __restrict__ S,    // (ROWS, NN)
    const float* __restrict__ E,    // (ROWS, DV)
    const float* __restrict__ A,    // (ROWS, DV)
    const float* __restrict__ M0,   // (NN, DV)
    float* __restrict__ R)          // (ROWS, DV)
{
  const int b     = blockIdx.x >> 2;
  const int chunk = blockIdx.x & 3;
  const int tid   = threadIdx.x;
  const int d     = tid >> 1;          // 0..63
  const int half  = tid & 1;           // n-half on adjacent lane
  const int dg    = chunk * 64 + d;
  const int n0    = half * 25;

  __shared__ float scs[2][NN];

  float mreg[25];
#pragma unroll
  for (int j = 0; j < 25; ++j) mreg[j] = M0[(n0 + j) * DV + dg];

  const size_t rowbase = (size_t)b * LL;
  for (int t = 0; t < LL; ++t) {
    const size_t row = rowbase + t;
    float* sb = scs[t & 1];
    if (tid < NN) sb[tid] = S[row * NN + tid];
    const float ed = E[row * DV + dg];
    const float ad = A[row * DV + dg];
    __syncthreads();

    float r = 0.0f;
#pragma unroll
    for (int j = 0; j < 25; ++j) {
      const float sn = sb[n0 + j];
      const float m  = mreg[j];
      r = fmaf(sn, m, r);                          // read (pre-update memory)
      mreg[j] = fmaf(sn, fmaf(-ed, m, ad), m);     // m + s*(a - e*m)
    }
    r += __shfl_xor(r, 1);                         // combine the two n-halves
    if (half == 0) R[row * DV + dg] = r;
  }
}

// ---------------------------------------------------------------------------
// Kernel 4: out = tanh([reads, problems] @ Wout^T + b). K = 384 split 256|128.
// One wave per 32x64 tile (8 accumulators), FP32 WMMA 16x16x4.
// ---------------------------------------------------------------------------
__global__ __launch_bounds__(256) void out_kernel(
    const float* __restrict__ Rd,   // (ROWS, DV)
    const float* __restrict__ P,    // (ROWS, DK)
    const float* __restrict__ WO,   // (HH, KCAT)
    const float* __restrict__ bo,   // (HH)
    float* __restrict__ OUT)        // (ROWS, HH)
{
  const int wave = blockIdx.x * 8 + (threadIdx.x >> 5);
  const int lane = threadIdx.x & 31;
  const int ct4 = wave & 3;            // 4 strips of 64 cols (HH/64)
  const int rt  = wave >> 2;           // 4096 row tiles of 32
  const int r0 = rt * 32, c0 = ct4 * 64;
  const int mn = lane & 15;
  const int kh = ((lane >> 4) & 1) * 2;

  const float* arowR = Rd + (size_t)(r0 + mn) * DV + kh;
  const float* arowP = P  + (size_t)(r0 + mn) * DK + kh;
  const float* brow  = WO + (size_t)(c0 + mn) * KCAT + kh;
  __builtin_prefetch(arowR, 0, 3);
  __builtin_prefetch(brow, 0, 3);

  v8f acc[2][4] = {};
#pragma unroll 2
  for (int k = 0; k < DV; k += 4) {    // reads part of concat
    v2f a0 = *(const v2f*)(arowR + k);
    v2f a1 = *(const v2f*)(arowR + 16 * DV + k);
    v2f b0 = *(const v2f*)(brow + 0 * 16 * KCAT + k);
    v2f b1 = *(const v2f*)(brow + 1 * 16 * KCAT + k);
    v2f b2 = *(const v2f*)(brow + 2 * 16 * KCAT + k);
    v2f b3 = *(const v2f*)(brow + 3 * 16 * KCAT + k);
    acc[0][0] = __builtin_amdgcn_wmma_f32_16x16x4_f32(false, a0, false, b0, (short)0, acc[0][0], false, false);
    acc[0][1] = __builtin_amdgcn_wmma_f32_16x16x4_f32(false, a0, false, b1, (short)0, acc[0][1], false, false);
    acc[0][2] = __builtin_amdgcn_wmma_f32_16x16x4_f32(false, a0, false, b2, (short)0, acc[0][2], false, false);
    acc[0][3] = __builtin_amdgcn_wmma_f32_16x16x4_f32(false, a0, false, b3, (short)0, acc[0][3], false, false);
    acc[1][0] = __builtin_amdgcn_wmma_f32_16x16x4_f32(false, a1, false, b0, (short)0, acc[1][0], false, false);
    acc[1][1] = __builtin_amdgcn_wmma_f32_16x16x4_f32(false, a1, false, b1, (short)0, acc[1][1], false, false);
    acc[1][2] = __builtin_amdgcn_wmma_f32_16x16x4_f32(false, a1, false, b2, (short)0, acc[1][2], false, false);
    acc[1][3] = __builtin_amdgcn_wmma_f32_16x16x4_f32(false, a1, false, b3, (short)0, acc[1][3], false, false);
  }
#pragma unroll 2
  for (int k = 0; k < DK; k += 4) {    // problems part of concat
    v2f a0 = *(const v2f*)(arowP + k);
    v2f a1 = *(const v2f*)(arowP + 16 * DK + k);
    v2f b0 = *(const v2f*)(brow + 0 * 16 * KCAT + DV + k);
    v2f b1 = *(const v2f*)(brow + 1 * 16 * KCAT + DV + k);
    v2f b2 = *(const v2f*)(brow + 2 * 16 * KCAT + DV + k);
    v2f b3 = *(const v2f*)(brow + 3 * 16 * KCAT + DV + k);
    acc[0][0] = __builtin_amdgcn_wmma_f32_16x16x4_f32(false, a0, false, b0, (short)0, acc[0][0], false, false);
    acc[0][1] = __builtin_amdgcn_wmma_f32_16x16x4_f32(false, a0, false, b1, (short)0, acc[0][1], false, false);
    acc[0][2] = __builtin_amdgcn_wmma_f32_16x16x4_f32(false, a0, false, b2, (short)0, acc[0][2], false, false);
    acc[0][3] = __builtin_amdgcn_wmma_f32_16x16x4_f32(false, a0, false, b3, (short)0, acc[0][3], false, false);
    acc[1][0] = __builtin_amdgcn_wmma_f32_16x16x4_f32(false, a1, false, b0, (short)0, acc[1][0], false, false);
    acc[1][1] = __builtin_amdgcn_wmma_f32_16x16x4_f32(false, a1, false, b1, (short)0, acc[1][1], false, false);
    acc[1][2] = __builtin_amdgcn_wmma_f32_16x16x4_f32(false, a1, false, b2, (short)0, acc[1][2], false, false);
    acc[1][3] = __builtin_amdgcn_wmma_f32_16x16x4_f32(false, a1, false, b3, (short)0, acc[1][3], false, false);
  }

#pragma unroll
  for (int rr = 0; rr < 2; ++rr) {
    const int rbase = r0 + rr * 16 + ((lane >> 4) & 1) * 8;
#pragma unroll
    for (int cc = 0; cc < 4; ++cc) {
      const int col = c0 + cc * 16 + mn;
      const float bb = bo[col];
#pragma unroll
      for (int i = 0; i < 8; ++i)
        OUT[(size_t)(rbase + i) * HH + col] = fast_tanh(acc[rr][cc][i] + bb);
    }
  }
}

// ---------------------------------------------------------------------------
extern "C" void kernel_launch(void* const* d_in, const int* in_sizes, int n_in,
                              void* d_out, int out_size, void* d_ws, size_t ws_size,
                              hipStream_t stream) {
  const float* problems     = (const float*)d_in[0];
  const float* interactions = (const float*)d_in[1];
  const float* w_key        = (const float*)d_in[2];
  const float* w_erase_w    = (const float*)d_in[3];
  const float* w_erase_b    = (const float*)d_in[4];
  const float* w_add_w      = (const float*)d_in[5];
  const float* w_add_b      = (const float*)d_in[6];
  const float* w_out_w      = (const float*)d_in[7];
  const float* w_out_b      = (const float*)d_in[8];
  const float* init_memory  = (const float*)d_in[9];
  float* out = (float*)d_out;

  float* e_ws = (float*)d_ws;
  float* a_ws = e_ws + (size_t)ROWS * DV;
  float* s_ws = a_ws + (size_t)ROWS * DV;
  float* r_ws = s_ws + (size_t)ROWS * NN;

  // gates: 4096 row-tiles(32) * 4 col-strips * 2 mats = 32768 waves / 8 per blk
  gates_kernel<<<4096, 256, 0, stream>>>(interactions, w_erase_w, w_erase_b,
                                         w_add_w, w_add_b, e_ws, a_ws);
  // score: one thread per row
  score_kernel<<<ROWS / 256, 256, 0, stream>>>(problems, w_key, s_ws);
  // scan: 128 batches * 4 d-chunks
  scan_kernel<<<BB * 4, 128, 0, stream>>>(s_ws, e_ws, a_ws, init_memory, r_ws);
  // output: 4096 row-tiles(32) * 4 col-strips = 16384 waves / 8 per block
  out_kernel<<<2048, 256, 0, stream>>>(r_ws, problems, w_out_w, w_out_b, out);
}